// ISTFT_18622978196167
// MI455X (gfx1250) — compile-verified
//
#include <hip/hip_runtime.h>

typedef __attribute__((ext_vector_type(2))) float v2f;
typedef __attribute__((ext_vector_type(8))) float v8f;

#define NFFT     2048
#define HOP      512
#define NBINS    1025     // NFFT/2 + 1
#define NFRAMES  1000
#define NBATCH   8
#define OUTLEN   511488   // HOP * (NFRAMES - 1)
#define NCOLS    (NBATCH * NFRAMES)   // 8000, GEMM N
#define KSTEPS   514      // 257 k-steps (K=4) real region + 257 imag region
#define MTILES   (NFFT / 16)          // 128
#define NTILES   (NCOLS / 16)         // 500
#define FLT_TINY 1.1754943508222875e-38f

// ---------------------------------------------------------------------------
// Fold Hermitian symmetry into the windowed-IDFT coefficients.
// K layout: k in [0,1028)   -> real-part coefs, f = k      (f >= 1025 -> 0)
//           k in [1028,2056)-> imag-part coefs, f = k-1028 (f >= 1025 -> 0)
// A[o,f] =  Wr[o,f] + (1<=f<=1023 ? Wr[o,2048-f] : 0)
// B[o,f] = -Wi[o,f] + (1<=f<=1023 ? Wi[o,2048-f] : 0)
// ---------------------------------------------------------------------------
__device__ __forceinline__ float fold_coef(const float* __restrict__ Wr,
                                           const float* __restrict__ Wi,
                                           int o, int k) {
  if (k < 1028) {
    int f = k;
    if (f >= NBINS) return 0.0f;
    float v = Wr[o * NFFT + f];
    if (f >= 1 && f <= NFFT / 2 - 1) v += Wr[o * NFFT + (NFFT - f)];
    return v;
  } else {
    int f = k - 1028;
    if (f >= NBINS) return 0.0f;
    float v = -Wi[o * NFFT + f];
    if (f >= 1 && f <= NFFT / 2 - 1) v += Wi[o * NFFT + (NFFT - f)];
    return v;
  }
}

// Pack folded coefficients into WMMA A-fragment order:
// apack[((mTile*KSTEPS + kk)*32 + lane)*2 + {0,1}] =
//   A2[o = mTile*16 + (lane&15)][k = kk*4 + 2*(lane>>4) + {0,1}]
__global__ void pack_a_kernel(const float* __restrict__ Wr,
                              const float* __restrict__ Wi,
                              float* __restrict__ apack) {
  int gid  = blockIdx.x * blockDim.x + threadIdx.x;
  int lane = gid & 31;
  int kk   = (gid >> 5) % KSTEPS;
  int mt   = (gid >> 5) / KSTEPS;
  if (mt >= MTILES) return;
  int o = mt * 16 + (lane & 15);
  int k = kk * 4 + 2 * (lane >> 4);
  int idx = ((mt * KSTEPS + kk) * 32 + lane) * 2;
  apack[idx + 0] = fold_coef(Wr, Wi, o, k);
  apack[idx + 1] = fold_coef(Wr, Wi, o, k + 1);
}

// ---------------------------------------------------------------------------
// WMMA GEMM: s[n, o] = sum_k A2[o,k] * X2[k,n]
//   X2[k, n] = real[n*1025 + k]        for k in [0,1025)
//            = imag[n*1025 + (k-1028)] for k in [1028,2053)
//            = 0                       elsewhere (pad)
// One wave computes a 64x64 tile: 4 M-tiles x 4 N-tiles, 16 accumulators,
// fp32 WMMA 16x16x4. Output written as s[b][o][t] (t contiguous stores).
// ---------------------------------------------------------------------------
#define LOADA(kk)                                                              \
  _Pragma("unroll") for (int i = 0; i < 4; ++i)                                \
      a[i] = Ab[((long)(mg4 + i) * KSTEPS + (kk)) * 32 + lane];

#define WMMA16()                                                               \
  _Pragma("unroll") for (int i = 0; i < 4; ++i) {                              \
    _Pragma("unroll") for (int j = 0; j < 4; ++j) {                            \
      c[i][j] = __builtin_amdgcn_wmma_f32_16x16x4_f32(                         \
          false, a[i], false, b[j], (short)0, c[i][j], false, false);          \
    }                                                                          \
  }

__global__ void __launch_bounds__(256)
istft_gemm_kernel(const float* __restrict__ xr, const float* __restrict__ xi,
                  const float* __restrict__ apack, float* __restrict__ sbuf) {
  const int lane = threadIdx.x & 31;
  const int w    = blockIdx.x * 8 + (threadIdx.x >> 5);
  const int mg   = w & 31;   // 32 groups of 4 M-tiles -> M = 2048
  const int ng   = w >> 5;   // 125 groups of 4 N-tiles -> N = 8000
  const int mg4  = mg * 4;
  const int h    = lane >> 4;  // half-wave selects K pair

  const v2f* Ab = (const v2f*)apack;

  // per-lane column start offsets for the 4 B fragments
  int colIdx[4];
#pragma unroll
  for (int j = 0; j < 4; ++j)
    colIdx[j] = (ng * 64 + j * 16 + (lane & 15)) * NBINS;

  v8f c[4][4];
#pragma unroll
  for (int i = 0; i < 4; ++i)
#pragma unroll
    for (int j = 0; j < 4; ++j) c[i][j] = (v8f){};

  v2f a[4], b[4];

  // --- region 0: real spectrum, kk = 0..255 (f = 4*kk + 2h, f+1 <= 1023) ---
#pragma unroll 2
  for (int kk = 0; kk < 256; ++kk) {
    const int f = kk * 4 + 2 * h;
    LOADA(kk);
#pragma unroll
    for (int j = 0; j < 4; ++j) {
      b[j].x = xr[colIdx[j] + f];
      b[j].y = xr[colIdx[j] + f + 1];
    }
    WMMA16();
  }
  { // kk = 256 edge: k = 1024 + 2h (only f = 1024 valid); branch-free select
    LOADA(256);
#pragma unroll
    for (int j = 0; j < 4; ++j) {
      float v = xr[colIdx[j] + 1024];  // in-bounds for all lanes
      b[j].x = (h == 0) ? v : 0.0f;
      b[j].y = 0.0f;
    }
    WMMA16();
  }

  // --- region 1: imag spectrum, kk = 257..512 (f = 4*(kk-257) + 2h) ---
#pragma unroll 2
  for (int kk = 257; kk < 513; ++kk) {
    const int f = (kk - 257) * 4 + 2 * h;
    LOADA(kk);
#pragma unroll
    for (int j = 0; j < 4; ++j) {
      b[j].x = xi[colIdx[j] + f];
      b[j].y = xi[colIdx[j] + f + 1];
    }
    WMMA16();
  }
  { // kk = 513 edge: f = 1024 + 2h (only f = 1024 valid)
    LOADA(513);
#pragma unroll
    for (int j = 0; j < 4; ++j) {
      float v = xi[colIdx[j] + 1024];
      b[j].x = (h == 0) ? v : 0.0f;
      b[j].y = 0.0f;
    }
    WMMA16();
  }

  // --- store D tiles: s[b][o][t], b = nn/1000, t = nn%1000 ---
  const int orow = h * 8;  // lanes 16..31 hold M = v + 8
#pragma unroll
  for (int j = 0; j < 4; ++j) {
    const int nn = ng * 64 + j * 16 + (lane & 15);
    const int bb = nn / NFRAMES;
    const int tt = nn - bb * NFRAMES;
    float* outb = sbuf + (long)bb * NFFT * NFRAMES + tt;
#pragma unroll
    for (int i = 0; i < 4; ++i) {
      const int ob = (mg4 + i) * 16 + orow;
#pragma unroll
      for (int v = 0; v < 8; ++v)
        outb[(long)(ob + v) * NFRAMES] = c[i][j][v];
    }
  }
}

// ---------------------------------------------------------------------------
// Overlap-add + window-sum-square normalization (gather form, deterministic).
// y[b, j] = sum_t s[b, j-512t, t] ; wss[j] = sum_t win[j-512t]^2 ; <=4 frames.
// Output slice: j = jo + 1024, jo in [0, OUTLEN).
// ---------------------------------------------------------------------------
__global__ void ola_kernel(const float* __restrict__ sbuf,
                           const float* __restrict__ win,
                           float* __restrict__ out) {
  int gid = blockIdx.x * blockDim.x + threadIdx.x;
  if (gid >= NBATCH * OUTLEN) return;
  int b  = gid / OUTLEN;
  int jo = gid - b * OUTLEN;
  int j  = jo + NFFT / 2;

  int tmax = j >> 9;  // j / HOP
  if (tmax > NFRAMES - 1) tmax = NFRAMES - 1;
  int tmin = (j >= NFFT - HOP) ? ((j - (NFFT - HOP)) >> 9) : 0;

  const float* sb = sbuf + (long)b * NFFT * NFRAMES;
  float acc = 0.0f, wss = 0.0f;
  for (int t = tmin; t <= tmax; ++t) {
    int o = j - (t << 9);
    acc += sb[(long)o * NFRAMES + t];
    float wv = win[o];
    wss += wv * wv;
  }
  out[gid] = (wss > FLT_TINY) ? acc / wss : acc;
}

// ---------------------------------------------------------------------------
extern "C" void kernel_launch(void* const* d_in, const int* in_sizes, int n_in,
                              void* d_out, int out_size, void* d_ws,
                              size_t ws_size, hipStream_t stream) {
  const float* real = (const float*)d_in[0];  // (8,1,1000,1025)
  const float* imag = (const float*)d_in[1];
  const float* Wr   = (const float*)d_in[2];  // (2048,2048) row-major [o][i]
  const float* Wi   = (const float*)d_in[3];
  const float* win  = (const float*)d_in[4];  // (2048,)
  float* out = (float*)d_out;                 // (8, 511488)

  float* apack = (float*)d_ws;                            // 16.84 MB
  size_t apack_elems = (size_t)MTILES * KSTEPS * 64;
  float* sbuf = apack + apack_elems;                      // 65.5 MB: s[b][o][t]

  {
    int total = MTILES * KSTEPS * 32;
    pack_a_kernel<<<(total + 255) / 256, 256, 0, stream>>>(Wr, Wi, apack);
  }
  // 32 m-groups x 125 n-groups = 4000 waves, 8 waves (256 thr) per block
  istft_gemm_kernel<<<(MTILES / 4) * (NTILES / 4) / 8, 256, 0, stream>>>(
      real, imag, apack, sbuf);
  {
    int total = NBATCH * OUTLEN;
    ola_kernel<<<(total + 255) / 256, 256, 0, stream>>>(sbuf, win, out);
  }
}